// PartitionedGraphAttentionLayer_67482526154914
// MI455X (gfx1250) — compile-verified
//
#include <hip/hip_runtime.h>
#include <math.h>

typedef __attribute__((ext_vector_type(2))) float v2f;
typedef __attribute__((ext_vector_type(8))) float v8f;

#define NB    32
#define CH    64
#define TDIM  300
#define VV    25
#define FF    64
#define EE    150
#define NPART 3
#define TBLK  4              // t-values per block
#define ROWS  100            // TBLK * VV rows of the fused GEMM
#define MPAD  112            // rows padded to 7 tiles of 16
#define HS    66             // LDS row stride (floats), conflict-free for D writes
#define TVPLANE 7500         // T*V

__launch_bounds__(256)
__global__ void gat_fused_kernel(const float* __restrict__ input,
                                 const float* __restrict__ W,
                                 const float* __restrict__ a,
                                 const int*   __restrict__ edge_index,
                                 const int*   __restrict__ edge_part,
                                 float*       __restrict__ out) {
  __shared__ float h_lds[MPAD * HS];       // h = x @ W  (112 x 64, stride 66)
  __shared__ float hp_lds[ROWS * HS];      // h' accumulator
  __shared__ float ssrc[ROWS * NPART];
  __shared__ float sdst[ROWS * NPART];
  __shared__ int   esrc[EE], edst[EE], epart[EE];
  __shared__ float e_lds[TBLK * EE];       // per-edge score -> exp -> alpha
  __shared__ float mseg[TBLK * 32];        // segment max
  __shared__ float dseg[TBLK * 32];        // segment sum

  const int tid  = threadIdx.x;
  const int lane = tid & 31;
  const int wv   = tid >> 5;
  const int blk  = blockIdx.x;
  const int n    = blk / (TDIM / TBLK);    // batch index
  const int tb   = blk % (TDIM / TBLK);    // t-chunk index
  const int tv0  = tb * ROWS;              // base flat (t*V+v) index, contiguous

  // ---- stage edge lists ----
  if (tid < EE) {
    esrc[tid]  = edge_index[tid];
    edst[tid]  = edge_index[EE + tid];
    epart[tid] = edge_part[tid];
  }
  for (int i = tid; i < ROWS * HS; i += 256) hp_lds[i] = 0.0f;

  // ---- GEMM h = A(112x64) * W(64x64) with V_WMMA_F32_16X16X4_F32 ----
  if (wv < 7) {
    const int  half = lane >> 4;           // 0: K=k0,k0+1   1: K=k0+2,k0+3
    const int  l15  = lane & 15;
    const int  mrow = wv * 16 + l15;       // A-fragment row (same for both halves)
    const bool mok  = (mrow < ROWS);
    const float* inb = input + (size_t)n * CH * TVPLANE + tv0 + mrow;

    v2f afrag[16];
#pragma unroll
    for (int kc = 0; kc < 16; ++kc) {      // preload entire A strip (read once)
      const int kb = kc * 4 + half * 2;
      afrag[kc].x = mok ? inb[(size_t)kb * TVPLANE]       : 0.0f;
      afrag[kc].y = mok ? inb[(size_t)(kb + 1) * TVPLANE] : 0.0f;
    }
#pragma unroll
    for (int nt = 0; nt < 4; ++nt) {
      const float* wb = W + nt * 16 + l15; // B fragment source (L2-resident)
      v8f acc = {};
#pragma unroll
      for (int kc = 0; kc < 16; ++kc) {
        const int kb = kc * 4 + half * 2;
        v2f b;
        b.x = wb[kb * FF];
        b.y = wb[(kb + 1) * FF];
        acc = __builtin_amdgcn_wmma_f32_16x16x4_f32(
                  false, afrag[kc], false, b, (short)0, acc, false, false);
      }
#pragma unroll
      for (int r = 0; r < 8; ++r) {        // D layout: VGPR r -> M = r + 8*half
        const int m  = wv * 16 + r + half * 8;
        const int nn = nt * 16 + l15;
        h_lds[m * HS + nn] = acc[r];
      }
    }
  }
  __syncthreads();

  // ---- attention scores: s_src/s_dst = h . a (3 partitions each) ----
  if (tid < ROWS) {
    const int r = tid;
#pragma unroll
    for (int p = 0; p < NPART; ++p) {
      const float* as = a + p * (2 * FF);
      const float* ad = as + FF;
      float s0 = 0.0f, s1 = 0.0f;
      for (int f = 0; f < FF; ++f) {
        const float hv = h_lds[r * HS + f];
        s0 += hv * as[f];
        s1 += hv * ad[f];
      }
      ssrc[r * NPART + p] = s0;
      sdst[r * NPART + p] = s1;
    }
  }
  __syncthreads();

  // ---- per-edge score + leaky relu (group g = t within block) ----
  {
    const int g = tid >> 6, gt = tid & 63;
    for (int e = gt; e < EE; e += 64) {
      const int sv = esrc[e], dv = edst[e], p = epart[e];
      float sc = ssrc[(g * VV + sv) * NPART + p] + sdst[(g * VV + dv) * NPART + p];
      e_lds[g * EE + e] = sc > 0.0f ? sc : 0.2f * sc;
    }
  }
  __syncthreads();

  // ---- segment max over dst (deterministic per-(g,v) scan) ----
  if (tid < TBLK * 32) {
    const int g = tid >> 5, v = tid & 31;
    if (v < VV) {
      float m = -3.4e38f;
      for (int e = 0; e < EE; ++e)
        if (edst[e] == v) { const float x = e_lds[g * EE + e]; m = x > m ? x : m; }
      mseg[g * 32 + v] = m;
    }
  }
  __syncthreads();

  // ---- exp(score - max) ----
  {
    const int g = tid >> 6, gt = tid & 63;
    for (int e = gt; e < EE; e += 64)
      e_lds[g * EE + e] = __expf(e_lds[g * EE + e] - mseg[g * 32 + edst[e]]);
  }
  __syncthreads();

  // ---- segment sum ----
  if (tid < TBLK * 32) {
    const int g = tid >> 5, v = tid & 31;
    if (v < VV) {
      float s = 0.0f;
      for (int e = 0; e < EE; ++e)
        if (edst[e] == v) s += e_lds[g * EE + e];
      dseg[g * 32 + v] = s;
    }
  }
  __syncthreads();

  // ---- alpha = exp / den ----
  {
    const int g = tid >> 6, gt = tid & 63;
    for (int e = gt; e < EE; e += 64)
      e_lds[g * EE + e] = e_lds[g * EE + e] / dseg[g * 32 + edst[e]];
  }
  __syncthreads();

  // ---- aggregation: thread = (g, f) owns column f of group g -> no races ----
  {
    const int g = tid >> 6, f = tid & 63;
    for (int e = 0; e < EE; ++e) {
      const int sv = esrc[e], dv = edst[e];
      hp_lds[(g * VV + dv) * HS + f] +=
          e_lds[g * EE + e] * h_lds[(g * VV + sv) * HS + f];
    }
  }
  __syncthreads();

  // ---- ELU + transposed store: out[n, f, t, v] ----
  {
    float* ob = out + (size_t)n * FF * TVPLANE + tv0;
    for (int i = tid; i < FF * ROWS; i += 256) {
      const int f = i / ROWS;
      const int r = i - f * ROWS;
      float x = hp_lds[r * HS + f];
      x = x > 0.0f ? x : expm1f(x);
      ob[(size_t)f * TVPLANE + r] = x;
    }
  }
}

extern "C" void kernel_launch(void* const* d_in, const int* in_sizes, int n_in,
                              void* d_out, int out_size, void* d_ws, size_t ws_size,
                              hipStream_t stream) {
  const float* input = (const float*)d_in[0];   // (32, 64, 300, 25) f32
  const float* W     = (const float*)d_in[1];   // (64, 64) f32
  const float* a     = (const float*)d_in[2];   // (3, 128) f32
  const int*   ei    = (const int*)  d_in[3];   // (2, 150) i32
  const int*   ep    = (const int*)  d_in[4];   // (150,) i32
  float*       out   = (float*)d_out;           // (32, 64, 300, 25) f32

  const int blocks = NB * (TDIM / TBLK);        // 32 * 75 = 2400
  gat_fused_kernel<<<blocks, 256, 0, stream>>>(input, W, a, ei, ep, out);
}